// GNN_9028021256834
// MI455X (gfx1250) — compile-verified
//
#include <hip/hip_runtime.h>
#include <hip/hip_bf16.h>

#define X_DIMS 256
#define Y_DIMS 64

typedef float v2f __attribute__((ext_vector_type(2)));
typedef float v8f __attribute__((ext_vector_type(8)));

// ---------------------------------------------------------------------------
// CSR row pointer: row[] is sorted, row_ptr[r] = lower_bound(row, r).
// ---------------------------------------------------------------------------
__global__ void build_rowptr_kernel(const int* __restrict__ row,
                                    int* __restrict__ row_ptr,
                                    int n_nodes, int n_edges) {
  int r = blockIdx.x * blockDim.x + threadIdx.x;
  if (r > n_nodes) return;
  int lo = 0, hi = n_edges;
  while (lo < hi) {
    int mid = (lo + hi) >> 1;
    if (row[mid] < r) lo = mid + 1; else hi = mid;
  }
  row_ptr[r] = lo;
}

// ---------------------------------------------------------------------------
// Pre-swizzle W [896,64] into WMMA B-operand order:
//   Wp[kstep][ntile][lane] = (W[4*kstep+2h][nt*16 + n], W[4*kstep+2h+1][...])
// with h = lane>>4, n = lane&15. One v2f per (kstep, ntile, lane) so the GEMM
// loop does a single coalesced b64 load per WMMA B operand.
// ---------------------------------------------------------------------------
__global__ void pack_w_kernel(const float* __restrict__ W, float* __restrict__ Wp,
                              int total) {  // total = (896/4)*4*32
  int t = blockIdx.x * blockDim.x + threadIdx.x;
  if (t >= total) return;
  int lane  = t & 31;
  int nt    = (t >> 5) & 3;
  int kstep = t >> 7;
  int h = lane >> 4;
  int n = (lane & 15) + nt * 16;
  int k = kstep * 4 + 2 * h;
  Wp[2 * t + 0] = W[(size_t)k * Y_DIMS + n];
  Wp[2 * t + 1] = W[(size_t)(k + 1) * Y_DIMS + n];
}

// ---------------------------------------------------------------------------
// Vectorized per-lane feature row load/store (b128 / b64 guaranteed).
// ---------------------------------------------------------------------------
template <int DPL>
__device__ __forceinline__ void load_vec(const float* __restrict__ p, float (&v)[DPL]) {
  if constexpr (DPL == 8) {
    float4 a = ((const float4*)p)[0];
    float4 b = ((const float4*)p)[1];
    v[0] = a.x; v[1] = a.y; v[2] = a.z; v[3] = a.w;
    v[4] = b.x; v[5] = b.y; v[6] = b.z; v[7] = b.w;
  } else {
    float2 a = ((const float2*)p)[0];
    v[0] = a.x; v[1] = a.y;
  }
}

template <int DPL>
__device__ __forceinline__ void store_vec(float* __restrict__ p, const float (&v)[DPL]) {
  if constexpr (DPL == 8) {
    float4 a = make_float4(v[0], v[1], v[2], v[3]);
    float4 b = make_float4(v[4], v[5], v[6], v[7]);
    ((float4*)p)[0] = a;
    ((float4*)p)[1] = b;
  } else {
    ((float2*)p)[0] = make_float2(v[0], v[1]);
  }
}

// ---------------------------------------------------------------------------
// Fused adaptive-weight propagation, one wave32 per node row.
//   out[r] = (sum_e exp(-||f_r - f_c||^2/s^2) * f_c) / (sum_e exp(...))
// Single pass per row: numerator and denominator accumulated together so each
// feats[col] row is gathered exactly once. Edge loop unrolled 2x for ILP
// (two independent load -> fma -> shfl-butterfly -> exp chains in flight).
// ---------------------------------------------------------------------------
template <int DPL>
__global__ void prop_kernel(const float* __restrict__ fin,
                            float* __restrict__ fout,
                            const int* __restrict__ col,
                            const int* __restrict__ row_ptr,
                            const float* __restrict__ sigmas, int sigma_idx,
                            int n_nodes) {
  const int D = 32 * DPL;
  int wave = (int)((blockIdx.x * blockDim.x + threadIdx.x) >> 5);
  int lane = threadIdx.x & 31;
  if (wave >= n_nodes) return;

  float s = sigmas[sigma_idx];
  float inv_s2 = 1.0f / (s * s);

  float fr[DPL], acc[DPL];
  load_vec<DPL>(fin + (size_t)wave * D + lane * DPL, fr);
#pragma unroll
  for (int i = 0; i < DPL; ++i) acc[i] = 0.0f;

  int e0 = row_ptr[wave];
  int e1 = row_ptr[wave + 1];
  float deg = 0.0f;

  int e = e0;
  for (; e + 1 < e1; e += 2) {
    int cA = col[e];
    int cB = col[e + 1];
    float fcA[DPL], fcB[DPL];
    load_vec<DPL>(fin + (size_t)cA * D + lane * DPL, fcA);
    load_vec<DPL>(fin + (size_t)cB * D + lane * DPL, fcB);

    float dA = 0.0f, dB = 0.0f;
#pragma unroll
    for (int i = 0; i < DPL; ++i) {
      float a = fr[i] - fcA[i];
      float b = fr[i] - fcB[i];
      dA = fmaf(a, a, dA);
      dB = fmaf(b, b, dB);
    }
#pragma unroll
    for (int off = 16; off >= 1; off >>= 1) {
      dA += __shfl_xor(dA, off, 32);
      dB += __shfl_xor(dB, off, 32);
    }
    float vA = __expf(-dA * inv_s2);
    float vB = __expf(-dB * inv_s2);
    deg += vA + vB;
#pragma unroll
    for (int i = 0; i < DPL; ++i)
      acc[i] = fmaf(vA, fcA[i], fmaf(vB, fcB[i], acc[i]));
  }
  if (e < e1) {
    int c = col[e];
    float fc[DPL];
    load_vec<DPL>(fin + (size_t)c * D + lane * DPL, fc);
    float d2 = 0.0f;
#pragma unroll
    for (int i = 0; i < DPL; ++i) {
      float d = fr[i] - fc[i];
      d2 = fmaf(d, d, d2);
    }
#pragma unroll
    for (int off = 16; off >= 1; off >>= 1) d2 += __shfl_xor(d2, off, 32);
    float val = __expf(-d2 * inv_s2);
    deg += val;
#pragma unroll
    for (int i = 0; i < DPL; ++i) acc[i] = fmaf(val, fc[i], acc[i]);
  }

  float inv = (deg > 0.0f) ? (1.0f / deg) : 0.0f;
  float outv[DPL];
#pragma unroll
  for (int i = 0; i < DPL; ++i) outv[i] = acc[i] * inv;
  store_vec<DPL>(fout + (size_t)wave * D + lane * DPL, outv);
}

// ---------------------------------------------------------------------------
// Final GEMM segment accumulator with V_WMMA_F32_16X16X4_F32 (fp32 exact).
// A 16x4 (ISA 7.12.2): lane half h -> K = 2h, 2h+1 (single b64 load).
// B comes pre-swizzled from Wp: one b64 load per WMMA, fully coalesced.
// ---------------------------------------------------------------------------
__device__ __forceinline__ void gemm_segment(const float* __restrict__ P, int Dseg,
                                             const v2f* __restrict__ Wp, int kstep0,
                                             int h, int lane,
                                             v8f& c0, v8f& c1, v8f& c2, v8f& c3) {
  for (int k = 0; k < Dseg; k += 4) {
    v2f a = *(const v2f*)(P + k + 2 * h);
    const v2f* Wk = Wp + (size_t)(kstep0 + (k >> 2)) * 128 + lane;
    c0 = __builtin_amdgcn_wmma_f32_16x16x4_f32(false, a, false, Wk[0],  (short)0, c0, false, false);
    c1 = __builtin_amdgcn_wmma_f32_16x16x4_f32(false, a, false, Wk[32], (short)0, c1, false, false);
    c2 = __builtin_amdgcn_wmma_f32_16x16x4_f32(false, a, false, Wk[64], (short)0, c2, false, false);
    c3 = __builtin_amdgcn_wmma_f32_16x16x4_f32(false, a, false, Wk[96], (short)0, c3, false, false);
  }
}

// One wave32 per 16-row tile; computes 16x64 of out = node_feats @ W walking
// K = 896 across the five concatenated segments (concat never materialized).
__global__ void gemm_kernel(const float* __restrict__ x,
                            const float* __restrict__ x1,
                            const float* __restrict__ x2,
                            const float* __restrict__ y1,
                            const float* __restrict__ y2,
                            const float* __restrict__ Wp_f,
                            float* __restrict__ out, int n_tiles) {
  int wave = (int)((blockIdx.x * blockDim.x + threadIdx.x) >> 5);
  int lane = threadIdx.x & 31;
  if (wave >= n_tiles) return;  // wave-uniform: EXEC all-ones for WMMA

  int m = lane & 15;
  int h = lane >> 4;
  const v2f* Wp = (const v2f*)Wp_f;

  size_t rowA = (size_t)wave * 16 + m;
  v8f c0 = {}, c1 = {}, c2 = {}, c3 = {};

  gemm_segment(x  + rowA * X_DIMS, X_DIMS, Wp,   0, h, lane, c0, c1, c2, c3);
  gemm_segment(x1 + rowA * X_DIMS, X_DIMS, Wp,  64, h, lane, c0, c1, c2, c3);
  gemm_segment(x2 + rowA * X_DIMS, X_DIMS, Wp, 128, h, lane, c0, c1, c2, c3);
  gemm_segment(y1 + rowA * Y_DIMS, Y_DIMS, Wp, 192, h, lane, c0, c1, c2, c3);
  gemm_segment(y2 + rowA * Y_DIMS, Y_DIMS, Wp, 208, h, lane, c0, c1, c2, c3);

  // C/D layout: VGPR j -> row (tile*16 + h*8 + j), col = ntile*16 + m
  float* O = out + ((size_t)wave * 16 + h * 8) * Y_DIMS + m;
#pragma unroll
  for (int j = 0; j < 8; ++j) {
    O[(size_t)j * Y_DIMS +  0] = c0[j];
    O[(size_t)j * Y_DIMS + 16] = c1[j];
    O[(size_t)j * Y_DIMS + 32] = c2[j];
    O[(size_t)j * Y_DIMS + 48] = c3[j];
  }
}

// ---------------------------------------------------------------------------
extern "C" void kernel_launch(void* const* d_in, const int* in_sizes, int n_in,
                              void* d_out, int out_size, void* d_ws, size_t ws_size,
                              hipStream_t stream) {
  (void)n_in; (void)out_size; (void)ws_size;
  const float* x    = (const float*)d_in[0];
  const float* y1h  = (const float*)d_in[1];
  const float* W    = (const float*)d_in[2];
  const float* sig  = (const float*)d_in[3];
  const int*   row  = (const int*)d_in[4];
  const int*   col  = (const int*)d_in[5];
  float*       out  = (float*)d_out;

  const int N = in_sizes[0] / X_DIMS;
  const int E = in_sizes[4];
  const int DIMS = X_DIMS * 3 + Y_DIMS * 2;          // 896
  const int WP_ELEMS = (DIMS / 4) * 4 * 32 * 2;      // packed W floats

  // workspace layout: x1, x2 [N,256]; y1, y2 [N,64]; row_ptr [N+1]; Wp
  float* x1 = (float*)d_ws;
  float* x2 = x1 + (size_t)N * X_DIMS;
  float* y1 = x2 + (size_t)N * X_DIMS;
  float* y2 = y1 + (size_t)N * Y_DIMS;
  int* row_ptr = (int*)(y2 + (size_t)N * Y_DIMS);
  float* Wp = (float*)(row_ptr + (N + 1) + 1);       // +1 pad keeps 8B align

  const int bs = 256;                                // 8 wave32s per block
  const int wpb = bs / 32;

  build_rowptr_kernel<<<(N + 1 + bs - 1) / bs, bs, 0, stream>>>(row, row_ptr, N, E);
  {
    int total = (DIMS / 4) * 4 * 32;
    pack_w_kernel<<<(total + bs - 1) / bs, bs, 0, stream>>>(W, Wp, total);
  }

  int grid = (N + wpb - 1) / wpb;                    // one wave per node row
  prop_kernel<8><<<grid, bs, 0, stream>>>(x,   x1, col, row_ptr, sig, 0, N);
  prop_kernel<8><<<grid, bs, 0, stream>>>(x1,  x2, col, row_ptr, sig, 1, N);
  prop_kernel<2><<<grid, bs, 0, stream>>>(y1h, y1, col, row_ptr, sig, 2, N);
  prop_kernel<2><<<grid, bs, 0, stream>>>(y1,  y2, col, row_ptr, sig, 3, N);

  int n_tiles = (N + 15) / 16;                       // N = 50000 -> 3125 tiles
  gemm_kernel<<<(n_tiles + wpb - 1) / wpb, bs, 0, stream>>>(x, x1, x2, y1, y2, Wp, out, n_tiles);
}